// QuantizedBn2d_72636486910332
// MI455X (gfx1250) — compile-verified
//
#include <hip/hip_runtime.h>

// QuantizedBn2d requant: memory-bound elementwise op (no matrix math -> no WMMA).
// 1 block per (b,c) plane (H*W = 3136 contiguous floats), wave-uniform scalar
// params, 128-bit non-temporal global loads/stores, gfx1250 speculative prefetch.
//
// multiply_M identity used:  multiply_M(sub, M0) == (sub*M0 + 2^30) >> 31 (arith),
// valid since the INT32_MIN overflow case is unreachable (|sub| <= 2^25, M0 > 0).

typedef float vfloat4 __attribute__((ext_vector_type(4)));

__device__ __forceinline__ float qmap(float xf, int A2, int C2, int m0,
                                      int spos, int mask, int halfm, int q3) {
    int xq  = __float2int_rn(xf);                 // v_cvt_i32_f32 (RTNE)
    int sub = xq * A2 + C2;                       // |sub| <= ~2^25, fits i32

    // multiply_M: single biased arithmetic shift (see identity above)
    long long prod = (long long)sub * (long long)m0;
    int m = (int)((prod + (1LL << 30)) >> 31);    // fits i32 (<= 2^25)

    // shifting4d: round-half-away-from-zero right shift by spos
    int rem = m & mask;
    int thr = halfm + (m < 0 ? 1 : 0);
    int res = (m >> spos) + (rem > thr ? 1 : 0) + q3;

    return (float)min(127, max(-128, res));       // v_med3_i32 + cvt
}

__global__ __launch_bounds__(256) void qbn2d_kernel(
    const float* __restrict__ x,
    const int*   __restrict__ bc,
    const int*   __restrict__ weight,
    const int*   __restrict__ bias,
    const int*   __restrict__ z1,
    const int*   __restrict__ z2,
    const int*   __restrict__ z3,
    const int*   __restrict__ M0,
    const int*   __restrict__ shift,
    float*       __restrict__ out,
    int C, int hw4)
{
    const int plane = blockIdx.x;          // b*C + c
    const int b = plane / C;
    const int c = plane - b * C;

    // Wave-uniform parameter gather (scalar loads: indexed only by blockIdx)
    const int k   = bc[b];
    const int w   = weight[k * C + c];
    const int bi  = bias[k * C + c];
    const int q1  = z1[k];
    const int q2  = z2[k];
    const int q3  = z3[k];
    const int m0  = M0[k];
    const int s   = shift[k];

    const int sneg = (s < 0) ? -s : 0;
    const int spos = (s > 0) ?  s : 0;
    // subsum = (xq - z1)*(w - z2) + bias ; pre-shift by sneg into constants
    const int A2    = (w - q2) << sneg;
    const int C2    = (bi << sneg) - q1 * A2;
    const int mask  = (1 << spos) - 1;
    const int halfm = mask >> 1;

    const vfloat4* __restrict__ xin =
        reinterpret_cast<const vfloat4*>(x) + (size_t)plane * hw4;
    vfloat4* __restrict__ po =
        reinterpret_cast<vfloat4*>(out) + (size_t)plane * hw4;

    const int step = blockDim.x;
    for (int i = threadIdx.x; i < hw4; i += step) {
        // gfx1250 speculative prefetch (TH=0): OOB translation drops silently,
        // so no bounds branch needed.
        __builtin_prefetch((const void*)&xin[i + step], 0, 0);

        // read-once stream: non-temporal b128 load
        vfloat4 v = __builtin_nontemporal_load(&xin[i]);

        vfloat4 r;
        r.x = qmap(v.x, A2, C2, m0, spos, mask, halfm, q3);
        r.y = qmap(v.y, A2, C2, m0, spos, mask, halfm, q3);
        r.z = qmap(v.z, A2, C2, m0, spos, mask, halfm, q3);
        r.w = qmap(v.w, A2, C2, m0, spos, mask, halfm, q3);

        // write-once stream: non-temporal b128 store
        __builtin_nontemporal_store(r, &po[i]);
    }
}

extern "C" void kernel_launch(void* const* d_in, const int* in_sizes, int n_in,
                              void* d_out, int out_size, void* d_ws, size_t ws_size,
                              hipStream_t stream) {
    const float* x      = (const float*)d_in[0];
    const int*   bc     = (const int*)d_in[1];
    const int*   weight = (const int*)d_in[2];
    const int*   bias   = (const int*)d_in[3];
    const int*   z1     = (const int*)d_in[4];
    const int*   z2     = (const int*)d_in[5];
    const int*   z3     = (const int*)d_in[6];
    const int*   M0     = (const int*)d_in[7];
    const int*   shift  = (const int*)d_in[8];
    float*       out    = (float*)d_out;

    const int B      = in_sizes[1];               // 64
    const int K      = in_sizes[4];               // 8 clusters
    const int C      = in_sizes[2] / K;           // 128
    const int planes = B * C;                     // 8192
    const int hw     = in_sizes[0] / planes;      // 3136 (divisible by 4)
    const int hw4    = hw / 4;                    // 784 float4 per plane

    qbn2d_kernel<<<planes, 256, 0, stream>>>(x, bc, weight, bias,
                                             z1, z2, z3, M0, shift,
                                             out, C, hw4);
}